// ASAP_79972291052239
// MI455X (gfx1250) — compile-verified
//
#include <hip/hip_runtime.h>

#define NG   64
#define NPG  2048
#define NN   131072            // NG * NPG
#define NE   1048576
#define UF   64
#define KT   512
#define ET   (NE + NN)         // edges + self loops
#define LEAK 0.2f

typedef __attribute__((ext_vector_type(2))) float v2f;
typedef __attribute__((ext_vector_type(8))) float v8f;

__device__ __forceinline__ void atomicMaxF(float* addr, float v) {
    // IEEE-ordered int trick: positives via signed max, negatives via unsigned min
    if (v >= 0.0f) atomicMax((int*)addr, __float_as_int(v));
    else           atomicMin((unsigned int*)addr, __float_as_uint(v));
}

__device__ __forceinline__ float leaky(float x) { return x > 0.0f ? x : LEAK * x; }

// ---------------- WMMA GEMM: [N,64] @ [64,64] (+bias), fp32 via v_wmma_f32_16x16x4_f32
__global__ void k_gemm64(const float* __restrict__ A, const float* __restrict__ W,
                         const float* __restrict__ bias, float* __restrict__ out) {
    __shared__ float Wl[64 * 64];
    int tid = threadIdx.x;                 // 256 threads = 8 waves
    for (int i = tid; i < 64 * 64; i += 256) Wl[i] = W[i];
    __syncthreads();
    int lane = tid & 31;
    int wave = tid >> 5;
    int row0 = blockIdx.x * 32 + (wave >> 2) * 16;   // 2 M-tiles
    int col0 = (wave & 3) * 16;                      // 4 N-tiles
    int mn   = lane & 15;
    int koff = (lane >> 4) * 2;                      // A/B 16x4 f32 layout: hi lanes hold K+2..3
    const float* arow = A + (size_t)(row0 + mn) * 64;
    v8f c = {};
#pragma unroll
    for (int kb = 0; kb < 64; kb += 4) {
        int kk = kb + koff;
        v2f a; a.x = arow[kk];                  a.y = arow[kk + 1];
        v2f b; b.x = Wl[kk * 64 + col0 + mn];   b.y = Wl[(kk + 1) * 64 + col0 + mn];
        c = __builtin_amdgcn_wmma_f32_16x16x4_f32(false, a, false, b, (short)0, c, false, false);
    }
    float bv = bias ? bias[col0 + mn] : 0.0f;
    int cm = (lane >> 4) * 8;                   // C/D: VGPR v holds M=v (lo lanes), M=v+8 (hi lanes)
#pragma unroll
    for (int v = 0; v < 8; ++v)
        out[(size_t)(row0 + cm + v) * 64 + col0 + mn] = c[v] + bv;
}

// ---------------- degree / dinv
__global__ void k_deg_init(float* deg) { deg[blockIdx.x * 256 + threadIdx.x] = 1.0f; }
__global__ void k_deg_edge(const int* __restrict__ ei, const float* __restrict__ ew, float* deg) {
    int e = blockIdx.x * 256 + threadIdx.x;
    atomicAdd(&deg[ei[e]], ew[e]);
}
__global__ void k_dinv(float* deg) {
    int n = blockIdx.x * 256 + threadIdx.x;
    float d = deg[n];
    deg[n] = d > 0.0f ? rsqrtf(d) : 0.0f;
}

// ---------------- attention GCN aggregation
__global__ void k_atth_init(const float* __restrict__ xw, const float* __restrict__ dinv,
                            const float* __restrict__ bias, float* atth) {
    int i = blockIdx.x * 256 + threadIdx.x;
    int n = i >> 6, u = i & 63;
    float di = dinv[n];                 // self loop: w=1, norm = dinv[n]^2
    atth[i] = di * di * xw[i] + bias[u];
}
__global__ void k_atth_edge(const int* __restrict__ ei, const float* __restrict__ ew,
                            const float* __restrict__ dinv, const float* __restrict__ xw,
                            float* atth) {
    int e = blockIdx.x * 4 + (threadIdx.x >> 6);
    int u = threadIdx.x & 63;
    int r = ei[e], c = ei[NE + e];
    float nw = dinv[r] * ew[e] * dinv[c];
    atomicAdd(&atth[r * 64 + u], nw * xw[c * 64 + u]);
}

// ---------------- segment max of neighbor features
__global__ void k_copy(const float* __restrict__ src, float* dst) {
    int i = blockIdx.x * 256 + threadIdx.x; dst[i] = src[i];
}
__global__ void k_max_edge(const int* __restrict__ ei, const float* __restrict__ atth, float* maxh) {
    int e = blockIdx.x * 4 + (threadIdx.x >> 6);
    int u = threadIdx.x & 63;
    int r = ei[e], c = ei[NE + e];
    atomicMaxF(&maxh[r * 64 + u], atth[c * 64 + u]);
}

// ---------------- score precomputation: qd = query.sk[:64], hd = atth.sk[64:]
__global__ void k_qh_dots(const float* __restrict__ query, const float* __restrict__ atth,
                          const float* __restrict__ sk, float* qd, float* hd) {
    int node = blockIdx.x * 8 + (threadIdx.x >> 5);  // wave32 per node
    int lane = threadIdx.x & 31;
    float2 q  = ((const float2*)(query + (size_t)node * 64))[lane];
    float2 h  = ((const float2*)(atth  + (size_t)node * 64))[lane];
    float2 k1 = ((const float2*)sk)[lane];
    float2 k2 = ((const float2*)(sk + 64))[lane];
    float s1 = q.x * k1.x + q.y * k1.y;
    float s2 = h.x * k2.x + h.y * k2.y;
    for (int m = 16; m > 0; m >>= 1) { s1 += __shfl_xor(s1, m, 32); s2 += __shfl_xor(s2, m, 32); }
    if (lane == 0) { qd[node] = s1; hd[node] = s2; }
}
__global__ void k_smax_init(const float* qd, const float* hd, const float* sb, float* smax) {
    int n = blockIdx.x * 256 + threadIdx.x;
    smax[n] = leaky(qd[n] + hd[n] + sb[0]);    // self-loop score
}
__global__ void k_score_edge(const int* __restrict__ ei, const float* __restrict__ qd,
                             const float* __restrict__ hd, const float* sb,
                             float* escore, float* smax) {
    int e = blockIdx.x * 256 + threadIdx.x;
    int r = ei[e], c = ei[NE + e];
    float s = leaky(qd[r] + hd[c] + sb[0]);
    escore[e] = s;
    atomicMaxF(&smax[r], s);
}
__global__ void k_sumexp_init(const float* qd, const float* hd, const float* sb,
                              const float* smax, float* sumexp) {
    int n = blockIdx.x * 256 + threadIdx.x;
    float s = leaky(qd[n] + hd[n] + sb[0]);
    sumexp[n] = expf(s - smax[n]);
}
__global__ void k_sumexp_edge(const int* __restrict__ ei, const float* __restrict__ escore,
                              const float* __restrict__ smax, float* sumexp) {
    int e = blockIdx.x * 256 + threadIdx.x;
    int r = ei[e];
    atomicAdd(&sumexp[r], expf(escore[e] - smax[r]));
}
__global__ void k_pool_init(const float* qd, const float* hd, const float* sb,
                            const float* smax, const float* sumexp,
                            const float* __restrict__ atth, float* pooled) {
    int i = blockIdx.x * 256 + threadIdx.x;
    int n = i >> 6;
    float s = leaky(qd[n] + hd[n] + sb[0]);
    float a = expf(s - smax[n]) / (sumexp[n] + 1e-16f);
    pooled[i] = a * atth[i];
}
__global__ void k_pool_edge(const int* __restrict__ ei, const float* __restrict__ escore,
                            const float* __restrict__ smax, const float* __restrict__ sumexp,
                            const float* __restrict__ atth, float* pooled) {
    int e = blockIdx.x * 4 + (threadIdx.x >> 6);
    int u = threadIdx.x & 63;
    int r = ei[e], c = ei[NE + e];
    float a = expf(escore[e] - smax[r]) / (sumexp[r] + 1e-16f);
    atomicAdd(&pooled[r * 64 + u], a * atth[c * 64 + u]);
}

// ---------------- LeConv fitness
__global__ void k_le_dots(const float* __restrict__ pooled, const float* ks, const float* kas,
                          const float* knb, const float* bs, const float* bas,
                          float* sself, float* sas, float* snb) {
    int node = blockIdx.x * 8 + (threadIdx.x >> 5);
    int lane = threadIdx.x & 31;
    float2 p = ((const float2*)(pooled + (size_t)node * 64))[lane];
    float2 a = ((const float2*)ks)[lane];
    float2 b = ((const float2*)kas)[lane];
    float2 c = ((const float2*)knb)[lane];
    float s1 = p.x * a.x + p.y * a.y;
    float s2 = p.x * b.x + p.y * b.y;
    float s3 = p.x * c.x + p.y * c.y;
    for (int m = 16; m > 0; m >>= 1) {
        s1 += __shfl_xor(s1, m, 32); s2 += __shfl_xor(s2, m, 32); s3 += __shfl_xor(s3, m, 32);
    }
    if (lane == 0) { sself[node] = s1 + bs[0]; sas[node] = s2 + bas[0]; snb[node] = s3; }
}
__global__ void k_aggr_init(const float* sas, const float* snb, float* aggr) {
    int n = blockIdx.x * 256 + threadIdx.x;
    aggr[n] = sas[n] - snb[n];                 // self loop, w = 1
}
__global__ void k_aggr_edge(const int* __restrict__ ei, const float* __restrict__ ew,
                            const float* __restrict__ sas, const float* __restrict__ snb,
                            float* aggr) {
    int e = blockIdx.x * 256 + threadIdx.x;
    int r = ei[e], c = ei[NE + e];
    atomicAdd(&aggr[r], ew[e] * (sas[r] - snb[c]));
}
__global__ void k_nscore(const float* sself, const float* aggr, float* nscore) {
    int n = blockIdx.x * 256 + threadIdx.x;
    float v = sself[n] + aggr[n];
    nscore[n] = 1.0f / (1.0f + expf(-v));
}

// ---------------- per-graph top-k via LDS bitonic sort (value desc, index asc)
__global__ void k_topk(const float* __restrict__ nscore, int* topk, float* outGidx) {
    __shared__ float sv[NPG];
    __shared__ int   si[NPG];
    int g = blockIdx.x;
    int tid = threadIdx.x;                     // 1024 threads
    for (int i = tid; i < NPG; i += 1024) { sv[i] = nscore[g * NPG + i]; si[i] = i; }
    __syncthreads();
    for (int k = 2; k <= NPG; k <<= 1) {
        for (int j = k >> 1; j > 0; j >>= 1) {
            for (int i = tid; i < NPG; i += 1024) {
                int ixj = i ^ j;
                if (ixj > i) {
                    float av = sv[i], bv = sv[ixj];
                    int   ai = si[i], bi = si[ixj];
                    bool p = (av > bv) || (av == bv && ai < bi);  // a precedes b
                    bool up = ((i & k) == 0);
                    bool dosw = up ? !p : p;
                    if (dosw) { sv[i] = bv; sv[ixj] = av; si[i] = bi; si[ixj] = ai; }
                }
            }
            __syncthreads();
        }
    }
    for (int j = tid; j < KT; j += 1024) {
        topk[g * KT + j]   = g * NPG + si[j];
        outGidx[g * KT + j] = (float)g;
    }
}

// ---------------- reindex + outputs
__global__ void k_newidx_init(int* newidx) { newidx[blockIdx.x * 256 + threadIdx.x] = -1; }
__global__ void k_scatter(const int* __restrict__ topk, int* newidx) {
    int r = blockIdx.x * 256 + threadIdx.x;
    newidx[topk[r]] = r;
}
__global__ void k_pooled_x(const int* __restrict__ topk, const float* __restrict__ pooled,
                           const float* __restrict__ nscore, float* outx) {
    int r = blockIdx.x * 4 + (threadIdx.x >> 6);
    int u = threadIdx.x & 63;
    int n = topk[r];
    outx[r * 64 + u] = pooled[n * 64 + u] * nscore[n];
}
__global__ void k_edges_out(const int* __restrict__ ei, const float* __restrict__ ew,
                            const int* __restrict__ newidx,
                            float* orow, float* ocol, float* ow) {
    int e = blockIdx.x * 256 + threadIdx.x;
    int r, c; float w;
    if (e < NE) { r = ei[e]; c = ei[NE + e]; w = ew[e]; }
    else        { r = c = e - NE; w = 1.0f; }
    int nr = newidx[r], nc = newidx[c];
    bool m = (nr >= 0) && (nc >= 0);
    orow[e] = (float)(m ? nr : -1);
    ocol[e] = (float)(m ? nc : -1);
    ow[e]   = m ? w : 0.0f;
}

extern "C" void kernel_launch(void* const* d_in, const int* in_sizes, int n_in,
                              void* d_out, int out_size, void* d_ws, size_t ws_size,
                              hipStream_t stream) {
    (void)in_sizes; (void)n_in; (void)out_size; (void)ws_size;
    const float* x   = (const float*)d_in[0];
    const int*   ei  = (const int*)  d_in[1];
    const float* ew  = (const float*)d_in[2];
    const float* Wg  = (const float*)d_in[4];
    const float* bg  = (const float*)d_in[5];
    const float* Wq  = (const float*)d_in[6];
    const float* bq  = (const float*)d_in[7];
    const float* sk  = (const float*)d_in[8];
    const float* sb  = (const float*)d_in[9];
    const float* ks  = (const float*)d_in[10];
    const float* bs  = (const float*)d_in[11];
    const float* kas = (const float*)d_in[12];
    const float* bas = (const float*)d_in[13];
    const float* knb = (const float*)d_in[14];

    char* p = (char*)d_ws;
    auto carve = [&](size_t bytes) { char* r = p; p += (bytes + 255) & ~(size_t)255; return r; };
    float* dinv   = (float*)carve((size_t)NN * 4);
    float* bufA   = (float*)carve((size_t)NN * UF * 4);  // xw -> query
    float* bufB   = (float*)carve((size_t)NN * UF * 4);  // attention_h
    float* bufC   = (float*)carve((size_t)NN * UF * 4);  // max_h -> pooled_h
    float* qd     = (float*)carve((size_t)NN * 4);
    float* hd     = (float*)carve((size_t)NN * 4);
    float* smax   = (float*)carve((size_t)NN * 4);
    float* sumexp = (float*)carve((size_t)NN * 4);
    float* escore = (float*)carve((size_t)NE * 4);
    float* sself  = (float*)carve((size_t)NN * 4);
    float* sas    = (float*)carve((size_t)NN * 4);
    float* snb    = (float*)carve((size_t)NN * 4);
    float* aggr   = (float*)carve((size_t)NN * 4);
    float* nscore = (float*)carve((size_t)NN * 4);
    int*   topk   = (int*)  carve((size_t)NG * KT * 4);
    int*   newidx = (int*)  carve((size_t)NN * 4);

    float* o_px  = (float*)d_out;                      // [NG*KT, 64]
    float* o_row = o_px + (size_t)NG * KT * UF;        // [ET]
    float* o_col = o_row + ET;                         // [ET]
    float* o_w   = o_col + ET;                         // [ET]
    float* o_gi  = o_w + ET;                           // [NG*KT]

    dim3 b256(256);
    // degree -> dinv
    k_deg_init<<<NN / 256, b256, 0, stream>>>(dinv);
    k_deg_edge<<<NE / 256, b256, 0, stream>>>(ei, ew, dinv);
    k_dinv<<<NN / 256, b256, 0, stream>>>(dinv);
    // xw = x @ Wgcn
    k_gemm64<<<NN / 32, b256, 0, stream>>>(x, Wg, nullptr, bufA);
    // attention_h
    k_atth_init<<<NN * UF / 256, b256, 0, stream>>>(bufA, dinv, bg, bufB);
    k_atth_edge<<<NE / 4, b256, 0, stream>>>(ei, ew, dinv, bufA, bufB);
    // max_h
    k_copy<<<NN * UF / 256, b256, 0, stream>>>(bufB, bufC);
    k_max_edge<<<NE / 4, b256, 0, stream>>>(ei, bufB, bufC);
    // query = max_h @ Wq + bq  (reuse bufA)
    k_gemm64<<<NN / 32, b256, 0, stream>>>(bufC, Wq, bq, bufA);
    // attention scores + segment softmax
    k_qh_dots<<<NN / 8, b256, 0, stream>>>(bufA, bufB, sk, qd, hd);
    k_smax_init<<<NN / 256, b256, 0, stream>>>(qd, hd, sb, smax);
    k_score_edge<<<NE / 256, b256, 0, stream>>>(ei, qd, hd, sb, escore, smax);
    k_sumexp_init<<<NN / 256, b256, 0, stream>>>(qd, hd, sb, smax, sumexp);
    k_sumexp_edge<<<NE / 256, b256, 0, stream>>>(ei, escore, smax, sumexp);
    // pooled_h (reuse bufC)
    k_pool_init<<<NN * UF / 256, b256, 0, stream>>>(qd, hd, sb, smax, sumexp, bufB, bufC);
    k_pool_edge<<<NE / 4, b256, 0, stream>>>(ei, escore, smax, sumexp, bufB, bufC);
    // LeConv fitness
    k_le_dots<<<NN / 8, b256, 0, stream>>>(bufC, ks, kas, knb, bs, bas, sself, sas, snb);
    k_aggr_init<<<NN / 256, b256, 0, stream>>>(sas, snb, aggr);
    k_aggr_edge<<<NE / 256, b256, 0, stream>>>(ei, ew, sas, snb, aggr);
    k_nscore<<<NN / 256, b256, 0, stream>>>(sself, aggr, nscore);
    // per-graph top-k + reindex
    k_topk<<<NG, dim3(1024), 0, stream>>>(nscore, topk, o_gi);
    k_newidx_init<<<NN / 256, b256, 0, stream>>>(newidx);
    k_scatter<<<NG * KT / 256, b256, 0, stream>>>(topk, newidx);
    // outputs
    k_pooled_x<<<NG * KT / 4, b256, 0, stream>>>(topk, bufC, nscore, o_px);
    k_edges_out<<<ET / 256, b256, 0, stream>>>(ei, ew, newidx, o_row, o_col, o_w);
}